// MembershipDecoder_13082470383826
// MI455X (gfx1250) — compile-verified
//
#include <hip/hip_runtime.h>
#include <stdint.h>

// L1-distance + row softmax:
//   dist[n,m] = sum_d |sites[n,d] - consensus[m,d]|,  out = softmax(-dist, axis=m)
// N=4096 sites, M=512 consensus rows, D=128 dims, all f32.
//
// MI455X strategy:
//  - No matmul structure in L1 distance -> VALU kernel at the 2-op/element floor
//    (v_sub_f32 + v_add_f32 with |src| modifier), dual-issue friendly.
//  - Whole consensus resident in WGP LDS (264 KB of 320 KB), row stride padded to
//    132 floats so the lane-strided ds_load_b128 reads are bank-conflict-free.
//  - LDS filled via CDNA5 async global->LDS copies (ASYNCcnt / s_wait_asynccnt),
//    issued from uniform-trip loops (no exec masking around the async ops).
//  - 512-thread blocks (16 wave32s), 32 site rows/block, 2 rows/wave; lane owns
//    m = lane + 32k -> coalesced stores; softmax via wave32 shuffle reductions.

#define D_DIM 128
#define M_DIM 512
#define ROW_PAD 132            // 128 + 4 floats padding (528 B row stride)
#define ROWS_PER_BLOCK 32
#define THREADS 512

__device__ __forceinline__ void async_copy16(uint32_t lds_byte_off, uint64_t gaddr) {
  // CDNA5: GLOBAL_LOAD_ASYNC_TO_LDS_B128 — per-lane 16B global->LDS DMA,
  // VDST vgpr holds the LDS byte address, tracked by ASYNCcnt.
  asm volatile("global_load_async_to_lds_b128 %0, %1, off"
               :: "v"(lds_byte_off), "v"(gaddr)
               : "memory");
}

__device__ __forceinline__ void wait_async0() {
#if __has_builtin(__builtin_amdgcn_s_wait_asynccnt)
  __builtin_amdgcn_s_wait_asynccnt(0);
#else
  asm volatile("s_wait_asynccnt 0" ::: "memory");
#endif
}

__global__ __launch_bounds__(THREADS)
void MembershipDecoder_l1softmax_kernel(const float* __restrict__ sites,
                                        const float* __restrict__ consensus,
                                        float* __restrict__ out,
                                        int n_sites) {
  extern __shared__ float lds[];
  // LDS layout: [0, M_DIM) padded consensus rows, then ROWS_PER_BLOCK padded site rows.
  // Low 32 bits of a generic LDS pointer == LDS byte offset on AMDGCN.
  const uint32_t ldsBase = (uint32_t)(uintptr_t)(void*)lds;
  const int tid    = threadIdx.x;
  const int block0 = blockIdx.x * ROWS_PER_BLOCK;

  // ---- Phase 1a: consensus -> LDS. 512 rows * 32 float4-chunks = 16384 transfers,
  // exactly 32 uniform iterations of 512 threads (no divergence, no bounds checks).
#pragma unroll 4
  for (int i = 0; i < (M_DIM * (D_DIM / 4)) / THREADS; ++i) {
    const int idx = i * THREADS + tid;
    const int row = idx >> 5;
    const int c   = idx & 31;
    const uint32_t dst = ldsBase + (uint32_t)(row * ROW_PAD + c * 4) * 4u;
    async_copy16(dst, (uint64_t)(uintptr_t)(consensus + (size_t)row * D_DIM + c * 4));
  }
  // ---- Phase 1b: this block's 32 site rows -> LDS. 1024 transfers = 2 iterations.
#pragma unroll
  for (int i = 0; i < (ROWS_PER_BLOCK * (D_DIM / 4)) / THREADS; ++i) {
    const int idx = i * THREADS + tid;
    const int row = idx >> 5;
    const int c   = idx & 31;
    const uint32_t dst = ldsBase + (uint32_t)((M_DIM + row) * ROW_PAD + c * 4) * 4u;
    async_copy16(dst, (uint64_t)(uintptr_t)(sites + (size_t)(block0 + row) * D_DIM + c * 4));
  }
  wait_async0();      // drain this wave's ASYNCcnt (34 outstanding max)
  __syncthreads();    // all waves' LDS writes visible

  // ---- Phase 2: L1 distances + softmax.
  const int wave = tid >> 5;
  const int lane = tid & 31;
  const float* consL = lds;
  const float* siteL = lds + M_DIM * ROW_PAD;

  for (int s = wave * 2; s < wave * 2 + 2; ++s) {
    float acc[16];
#pragma unroll
    for (int k = 0; k < 16; ++k) acc[k] = 0.0f;

    const float4* srow = (const float4*)(siteL + s * ROW_PAD);
#pragma unroll 2
    for (int c = 0; c < D_DIM / 4; ++c) {
      const float4 sv = srow[c];                       // broadcast LDS read
#pragma unroll
      for (int k = 0; k < 16; ++k) {
        const int m = lane + (k << 5);
        const float4 cv = *(const float4*)(consL + m * ROW_PAD + c * 4);
        // 2 VALU per element: v_sub_f32 + v_add_f32 with |src| modifier;
        // 16 independent acc chains -> VOPD dual-issue fodder.
        acc[k] += fabsf(sv.x - cv.x);
        acc[k] += fabsf(sv.y - cv.y);
        acc[k] += fabsf(sv.z - cv.z);
        acc[k] += fabsf(sv.w - cv.w);
      }
    }

    // Softmax over 512 logits (-acc), 16 per lane + wave32 shuffle reductions.
    float mx = -acc[0];
#pragma unroll
    for (int k = 1; k < 16; ++k) mx = fmaxf(mx, -acc[k]);
#pragma unroll
    for (int off = 16; off >= 1; off >>= 1)
      mx = fmaxf(mx, __shfl_xor(mx, off, 32));

    float e[16];
    float sum = 0.0f;
#pragma unroll
    for (int k = 0; k < 16; ++k) { e[k] = __expf(-acc[k] - mx); sum += e[k]; }
#pragma unroll
    for (int off = 16; off >= 1; off >>= 1)
      sum += __shfl_xor(sum, off, 32);

    const float inv = 1.0f / sum;
    float* orow = out + (size_t)(block0 + s) * M_DIM;
#pragma unroll
    for (int k = 0; k < 16; ++k)
      orow[lane + (k << 5)] = e[k] * inv;   // coalesced 128B stores per wave
  }
}

extern "C" void kernel_launch(void* const* d_in, const int* in_sizes, int n_in,
                              void* d_out, int out_size, void* d_ws, size_t ws_size,
                              hipStream_t stream) {
  const float* sites     = (const float*)d_in[0];
  const float* consensus = (const float*)d_in[1];
  float* out             = (float*)d_out;

  const int N = in_sizes[0] / D_DIM;             // 4096
  dim3 grid(N / ROWS_PER_BLOCK);                 // 128 blocks
  dim3 block(THREADS);                           // 512 threads = 16 wave32s
  const size_t lds_bytes =
      (size_t)(M_DIM + ROWS_PER_BLOCK) * ROW_PAD * sizeof(float); // ~280 KB < 320 KB

  MembershipDecoder_l1softmax_kernel<<<grid, block, lds_bytes, stream>>>(
      sites, consensus, out, N);
}